// GINELayer_21801253994645
// MI455X (gfx1250) — compile-verified
//
#include <hip/hip_runtime.h>
#include <hip/hip_bf16.h>

#define D 128
#define G 512
#define WAVES 8
#define BLOCK (WAVES * 32)
#define GN_EPS 1e-5f

typedef float v2f __attribute__((ext_vector_type(2)));
typedef float v8f __attribute__((ext_vector_type(8)));

static __device__ __forceinline__ v8f wmma_f32(v2f a, v2f b, v8f c) {
    // V_WMMA_F32_16X16X4_F32 : D = A(16x4 f32) * B(4x16 f32) + C(16x16 f32)
    return __builtin_amdgcn_wmma_f32_16x16x4_f32(false, a, false, b, (short)0, c,
                                                 false, false);
}

static __device__ __forceinline__ float silu_f(float x) {
    return x / (1.0f + __expf(-x));
}

// Cooperatively stage the 128x128 weight into LDS with a per-row rotation
// swizzle (row r rotated by 4*r floats) so B-fragment ds_load_b64 across a
// wave touches 64 distinct banks. Total = exactly 64 KB.
static __device__ __forceinline__ void load_W_lds(const float* __restrict__ W,
                                                  float* lds) {
    for (int i = threadIdx.x; i < D * (D / 4); i += BLOCK) {
        int r  = i >> 5;     // row (0..127)
        int c4 = i & 31;     // float4 index within row
        float4 v = ((const float4*)W)[i];
        int pos = ((c4 * 4) + (r * 4)) & (D - 1);
        *(float4*)&lds[r * D + pos] = v;
    }
}

// One wave computes a 16-row x 128-col fp32 tile: out = A[m0:m0+16,:] @ W^T + bias.
// A-fragment (ISA 7.12.2, 32-bit A 16x4): lane l<16 -> row m0+l, K={4k,4k+1};
// lane l>=16 -> row m0+l-16, K={4k+2,4k+3}.  B = W^T so B[k][n] = W[n][k]:
// per-lane float2 from W row (tile*16 + l&15) at the same K offset.
static __device__ __forceinline__ void gemm_rows16(const float* __restrict__ A,
                                                   long m0, const float* lds,
                                                   const float* __restrict__ bias,
                                                   int lane, v8f acc[8]) {
    const int lo = lane & 15;
    const int hi = lane >> 4;
#pragma unroll
    for (int nt = 0; nt < 8; ++nt) {
        float bv = bias[nt * 16 + lo];
        v8f c;
#pragma unroll
        for (int v = 0; v < 8; ++v) c[v] = bv;
        acc[nt] = c;
    }
    const float* arow = A + (m0 + lo) * D + 2 * hi;
#pragma unroll
    for (int ks = 0; ks < D / 4; ++ks) {
        const int coff = ks * 4 + 2 * hi;
        v2f a = *(const v2f*)(arow + ks * 4);
#pragma unroll
        for (int nt = 0; nt < 8; ++nt) {
            const int r = nt * 16 + lo;
            v2f b = *(const v2f*)&lds[r * D + ((coff + 4 * r) & (D - 1))];
            acc[nt] = wmma_f32(a, b, acc[nt]);
        }
    }
}

// dx = x @ W1^T + b1
__global__ __launch_bounds__(BLOCK) void node_gemm_k(const float* __restrict__ x,
                                                     const float* __restrict__ W1,
                                                     const float* __restrict__ b1,
                                                     float* __restrict__ dx,
                                                     int nRowBlocks) {
    __shared__ float lds[D * D];
    load_W_lds(W1, lds);
    __syncthreads();
    const int wave = threadIdx.x >> 5;
    const int lane = threadIdx.x & 31;
    const int rb = blockIdx.x * WAVES + wave;
    if (rb >= nRowBlocks) return;  // wave-uniform: EXEC stays all-ones for WMMA
    v8f acc[8];
    const long m0 = (long)rb * 16;
    gemm_rows16(x, m0, lds, b1, lane, acc);
    const int lo = lane & 15, hi = lane >> 4;
#pragma unroll
    for (int nt = 0; nt < 8; ++nt) {
        const int col = nt * 16 + lo;
#pragma unroll
        for (int v = 0; v < 8; ++v)
            dx[(m0 + 8 * hi + v) * D + col] = acc[nt][v];
    }
}

// ef = e @ Wl^T + bl ; msg = silu(dx[src] + ef) ; agg[dst] += msg
__global__ __launch_bounds__(BLOCK) void edge_gemm_scatter_k(
    const float* __restrict__ e, const float* __restrict__ Wl,
    const float* __restrict__ bl, const float* __restrict__ dx,
    const int* __restrict__ src, const int* __restrict__ dst,
    float* __restrict__ agg, int nRowBlocks) {
    __shared__ float lds[D * D];
    load_W_lds(Wl, lds);
    __syncthreads();
    const int wave = threadIdx.x >> 5;
    const int lane = threadIdx.x & 31;
    const int rb = blockIdx.x * WAVES + wave;
    if (rb >= nRowBlocks) return;
    v8f acc[8];
    const long m0 = (long)rb * 16;
    gemm_rows16(e, m0, lds, bl, lane, acc);
    const int lo = lane & 15, hi = lane >> 4;
    const long eb = m0 + 8 * hi;
    int sv[8], dv[8];
#pragma unroll
    for (int v = 0; v < 8; ++v) {
        sv[v] = src[eb + v];
        dv[v] = dst[eb + v];
    }
#pragma unroll
    for (int nt = 0; nt < 8; ++nt) {
        const int col = nt * 16 + lo;
#pragma unroll
        for (int v = 0; v < 8; ++v) {
            float m = acc[nt][v] + dx[(long)sv[v] * D + col];  // L2-resident gather
            m = silu_f(m);
            atomicAdd(&agg[(long)dv[v] * D + col], m);         // L2-resident scatter
        }
    }
}

// h = x + silu((1+eps)*dx + agg); out = h; per-graph sum & count
__global__ void node_update_k(const float* __restrict__ x,
                              const float* __restrict__ dx,
                              const float* __restrict__ agg,
                              const float* __restrict__ epsp,
                              const int* __restrict__ batch,
                              float* __restrict__ out, float* __restrict__ gsum,
                              float* __restrict__ gcnt, int N) {
    long idx = (long)blockIdx.x * blockDim.x + threadIdx.x;
    if (idx >= (long)N * D) return;
    const int n = (int)(idx >> 7);
    const int d = (int)(idx & (D - 1));
    float v = (1.0f + epsp[0]) * dx[idx] + agg[idx];
    v = silu_f(v);
    float h = x[idx] + v;
    out[idx] = h;
    const int g = batch[n];
    atomicAdd(&gsum[g * D + d], h);
    if (d == 0) atomicAdd(&gcnt[g], 1.0f);
}

__global__ void mean_k(float* __restrict__ gsum, const float* __restrict__ gcnt) {
    int idx = blockIdx.x * blockDim.x + threadIdx.x;
    if (idx >= G * D) return;
    float c = fmaxf(gcnt[idx >> 7], 1.0f);
    gsum[idx] = gsum[idx] / c;  // gsum becomes mean
}

// t = h - mean[batch]*mean_scale ; out = t ; per-graph sum of t^2
__global__ void center_k(float* __restrict__ out, const float* __restrict__ mean,
                         const float* __restrict__ gms,
                         const int* __restrict__ batch,
                         float* __restrict__ gsumsq, int N) {
    long idx = (long)blockIdx.x * blockDim.x + threadIdx.x;
    if (idx >= (long)N * D) return;
    const int n = (int)(idx >> 7);
    const int d = (int)(idx & (D - 1));
    const int g = batch[n];
    float t = out[idx] - mean[g * D + d] * gms[d];
    out[idx] = t;
    atomicAdd(&gsumsq[g * D + d], t * t);
}

__global__ void invstd_k(float* __restrict__ gsumsq,
                         const float* __restrict__ gcnt) {
    int idx = blockIdx.x * blockDim.x + threadIdx.x;
    if (idx >= G * D) return;
    float c = fmaxf(gcnt[idx >> 7], 1.0f);
    gsumsq[idx] = rsqrtf(gsumsq[idx] / c + GN_EPS);  // gsumsq becomes 1/std
}

__global__ void final_k(float* __restrict__ out, const float* __restrict__ istd,
                        const float* __restrict__ gw, const float* __restrict__ gb,
                        const int* __restrict__ batch, int N) {
    long idx = (long)blockIdx.x * blockDim.x + threadIdx.x;
    if (idx >= (long)N * D) return;
    const int n = (int)(idx >> 7);
    const int d = (int)(idx & (D - 1));
    const int g = batch[n];
    out[idx] = gw[d] * out[idx] * istd[g * D + d] + gb[d];
}

extern "C" void kernel_launch(void* const* d_in, const int* in_sizes, int n_in,
                              void* d_out, int out_size, void* d_ws, size_t ws_size,
                              hipStream_t stream) {
    const float* x   = (const float*)d_in[0];
    const float* e   = (const float*)d_in[1];
    const float* W1  = (const float*)d_in[2];
    const float* b1  = (const float*)d_in[3];
    const float* Wl  = (const float*)d_in[4];
    const float* bl  = (const float*)d_in[5];
    const float* eps = (const float*)d_in[6];
    const float* gw  = (const float*)d_in[7];
    const float* gb  = (const float*)d_in[8];
    const float* gms = (const float*)d_in[9];
    const int* batch = (const int*)d_in[10];
    const int* eidx  = (const int*)d_in[11];

    const int N = in_sizes[0] / D;   // 100000
    const int E = in_sizes[1] / D;   // 600000
    const int* src = eidx;           // edge_index[0]
    const int* dst = eidx + E;       // edge_index[1]
    float* out = (float*)d_out;

    // Workspace layout (floats): [agg N*D][gsum G*D][gsumsq G*D][gcnt G][dx N*D]
    float* ws     = (float*)d_ws;
    float* agg    = ws;
    float* gsum   = agg + (size_t)N * D;
    float* gsumsq = gsum + (size_t)G * D;
    float* gcnt   = gsumsq + (size_t)G * D;
    float* dx     = gcnt + G;

    const size_t zeroFloats = (size_t)N * D + 2 * (size_t)G * D + G;
    hipMemsetAsync(agg, 0, zeroFloats * sizeof(float), stream);

    const int nbN = N / 16, nbE = E / 16;
    node_gemm_k<<<(nbN + WAVES - 1) / WAVES, BLOCK, 0, stream>>>(x, W1, b1, dx, nbN);
    edge_gemm_scatter_k<<<(nbE + WAVES - 1) / WAVES, BLOCK, 0, stream>>>(
        e, Wl, bl, dx, src, dst, agg, nbE);

    const long ND = (long)N * D;
    const int tpb = 256;
    const int gND = (int)((ND + tpb - 1) / tpb);
    node_update_k<<<gND, tpb, 0, stream>>>(x, dx, agg, eps, batch, out, gsum, gcnt, N);
    mean_k<<<(G * D) / tpb, tpb, 0, stream>>>(gsum, gcnt);
    center_k<<<gND, tpb, 0, stream>>>(out, gsum, gms, batch, gsumsq, N);
    invstd_k<<<(G * D) / tpb, tpb, 0, stream>>>(gsumsq, gcnt);
    final_k<<<gND, tpb, 0, stream>>>(out, gsumsq, gw, gb, batch, N);
}